// CornerProposal_11330123726922
// MI455X (gfx1250) — compile-verified
//
#include <hip/hip_runtime.h>
#include <hip/hip_bf16.h>
#include <stdint.h>

// Problem constants (fixed by the reference's setup_inputs()).
namespace {
constexpr int kB = 8, kN = 600, kC = 32, kH = 256, kW = 256;
constexpr int kG  = 31;                 // glimpse size
constexpr int kGP = kG * kG;            // 961 pixels per glimpse per channel
constexpr int kCG = 16;                 // channels staged per LDS group
constexpr int kRow = 40;                // staged row width (floats), 16B-aligned start
constexpr int kWinRows = 32;            // staged rows per channel
constexpr int kBufFloats = kCG * kWinRows * kRow;         // 20480 floats = 80 KB / buffer
constexpr size_t kRoisElems = (size_t)kB * kN * kC * kGP; // 147,609,600
}

typedef unsigned int u32x4 __attribute__((ext_vector_type(4)));
typedef unsigned int u32x8 __attribute__((ext_vector_type(8)));

// ---------------------------------------------------------------------------
// Tensor Data Mover: one instruction stages a 3D tile (kRow x kWinRows x zext)
// from global (row stride W, plane stride H*W) into contiguous LDS.
// D# layout per CDNA5 ISA. Tracked by TENSORcnt.
// ---------------------------------------------------------------------------
__device__ __forceinline__ void tdm_load_3d_to_lds(unsigned lds_byte,
                                                   const float* src,
                                                   unsigned zextent) {
    const uint64_t ga = (uint64_t)(uintptr_t)src;
    const u32x4 g0 = {
        1u,                                                    // count=1, user desc
        lds_byte,                                              // lds_addr
        (unsigned)ga,                                          // global_addr[31:0]
        ((unsigned)(ga >> 32) & 0x01FFFFFFu) | (2u << 30)      // addr[56:32] | type=2
    };
    const u32x8 g1 = {
        2u << 16,                          // wg_mask=0, data_size=2 (4 bytes)
        (unsigned)kW << 16,                // tensor_dim0[15:0] @ bits63:48  (=W)
        (unsigned)kH << 16,                // dim0 hi=0, tensor_dim1[15:0]   (=H)
        (unsigned)kRow << 16,              // dim1 hi=0, tile_dim0 = 40
        (unsigned)kWinRows | (zextent << 16),  // tile_dim1=32, tile_dim2=zext
        (unsigned)kW,                      // tensor_dim0_stride lo32 (=W)
        0u,                                // dim0_stride hi16=0, dim1_stride lo16 (65536&0xffff=0)
        (unsigned)((kH * kW) >> 16)        // tensor_dim1_stride bits47:16 (=1 -> 65536)
    };
    const u32x4 g2 = { zextent, 0u, 0u, 0u };  // tensor_dim2; no dim3, not iterating
    const u32x4 g3 = { 0u, 0u, 0u, 0u };
    asm volatile("tensor_load_to_lds %0, %1, %2, %3"
                 :
                 : "s"(g0), "s"(g1), "s"(g2), "s"(g3)
                 : "memory");
}

// CDNA5 per-lane async global->LDS copy (ASYNCcnt-tracked), 16 bytes per lane.
__device__ __forceinline__ void async_copy_b128(unsigned lds_byte_addr, const float* src) {
    asm volatile("global_load_async_to_lds_b128 %0, %1, off"
                 :
                 : "v"(lds_byte_addr), "v"((unsigned long long)(uintptr_t)src)
                 : "memory");
}

// Stage a 16-channel x 32-row x 40-float aligned window into LDS (per-lane async).
__device__ __forceinline__ void stage_group_async(const float* __restrict__ images,
                                                  int b, int cb, int oy, int ox_al,
                                                  float* buf) {
    const unsigned base = (unsigned)(uintptr_t)buf;
    for (int i = threadIdx.x; i < kCG * kWinRows * (kRow / 4); i += 256) {
        int c   = i / (kWinRows * (kRow / 4));
        int rem = i - c * (kWinRows * (kRow / 4));
        int y   = rem / (kRow / 4);
        int v   = rem - y * (kRow / 4);
        const float* src = images
            + ((((size_t)b * kC + cb + c) * kH) + (size_t)(oy + y)) * kW
            + ox_al + v * 4;
        unsigned dst = base + (unsigned)(((c * kWinRows + y) * kRow + v * 4) * 4);
        async_copy_b128(dst, src);
    }
}

// Average the 2x2 block for every output pixel of this channel group.
// Non-temporal stores keep the 64 MB image set resident in L2 under the
// 590 MB output write stream.
__device__ __forceinline__ void compute_group(const float* buf, int shift,
                                              float* __restrict__ rois,
                                              size_t bn, int cb) {
    for (int o = threadIdx.x; o < kCG * kGP; o += 256) {
        int c  = o / kGP;
        int p  = o - c * kGP;
        int gy = p / kG;
        int gx = p - gy * kG;
        const float* w = buf + (c * kWinRows + gy) * kRow + gx + shift;
        float v00 = w[0], v01 = w[1], v10 = w[kRow], v11 = w[kRow + 1];
        float r = v00 * 0.25f + v01 * 0.25f + v10 * 0.25f + v11 * 0.25f;
        __builtin_nontemporal_store(r, &rois[(bn * kC + cb + c) * kGP + p]);
    }
}

__device__ __forceinline__ float tap(const float* img, int y, int x) {
    if (y < 0 || y >= kH || x < 0 || x >= kW) return 0.0f;
    return img[(size_t)y * kW + x];
}

__global__ void __launch_bounds__(256)
corner_glimpse_kernel(const float* __restrict__ images,
                      const float* __restrict__ anc,
                      float* __restrict__ rois,
                      float* __restrict__ anc_out) {
    extern __shared__ float smem[];    // 2 * 80 KB double buffer
    const int bn = blockIdx.x;         // b*N + n
    const int b  = bn / kN;

    // Anchor xy1 (uniform per block -> scalar loads).
    const float ax = anc[(size_t)bn * 4 + 0];
    const float ay = anc[(size_t)bn * 4 + 1];
    if (threadIdx.x == 0) {
        anc_out[(size_t)bn * 2 + 0] = ax;
        anc_out[(size_t)bn * 2 + 1] = ay;
    }

    // centers = floor(anchor); sample offset -0.5 => exact half-pel bilinear
    // = 2x2 box average over window rows/cols [c-16, c+15].
    const int cx = (int)floorf(ax);
    const int cy = (int)floorf(ay);
    const int ox = cx - 16, oy = cy - 16;
    const int ox_al = ox & ~3;         // 16B-aligned staging column
    const int shift = ox - ox_al;      // 0..3

    const bool fast = (ox >= 0) && (oy >= 0) &&
                      (ox_al + kRow <= kW) && (oy + kWinRows <= kH);
    if (fast) {
        float* buf0 = smem;
        float* buf1 = smem + kBufFloats;

        // Group 0 (channels 0..15): each of the 8 waves issues one TDM
        // descriptor for a 2-channel slice. TDM ignores EXEC, so the slice id
        // comes from a wave-uniform scalar (readfirstlane of tid>>5).
        {
            const int wid = __builtin_amdgcn_readfirstlane((int)(threadIdx.x >> 5));
            const float* src0 = images
                + (((size_t)b * kC + (size_t)(wid * 2)) * kH + (size_t)oy) * kW + ox_al;
            const unsigned lds0 = (unsigned)(uintptr_t)buf0
                + (unsigned)(wid * 2 * kWinRows * kRow * 4);
            tdm_load_3d_to_lds(lds0, src0, 2u);
        }

        // Group 1 (channels 16..31): per-lane async b128 path, in flight
        // concurrently with the TDM transfers and group-0 compute.
        stage_group_async(images, b, kCG, oy, ox_al, buf1);

        __builtin_amdgcn_s_wait_tensorcnt(0);             // own TDM slice done
        __syncthreads();                                  // all slices visible
        compute_group(buf0, shift, rois, (size_t)bn, 0);

        __builtin_amdgcn_s_wait_asynccnt(0);              // own async loads done
        __syncthreads();
        compute_group(buf1, shift, rois, (size_t)bn, kCG);
    } else {
        // General zero-padded bilinear fallback (never taken for these inputs).
        const float* imgb = images + (size_t)b * kC * kH * kW;
        for (int o = threadIdx.x; o < kC * kGP; o += 256) {
            int c  = o / kGP;
            int p  = o - c * kGP;
            int gy = p / kG;
            int gx = p - gy * kG;
            const float* img = imgb + (size_t)c * kH * kW;
            int y0 = cy + gy - 16, x0 = cx + gx - 16;
            float r = tap(img, y0,     x0    ) * 0.25f
                    + tap(img, y0,     x0 + 1) * 0.25f
                    + tap(img, y0 + 1, x0    ) * 0.25f
                    + tap(img, y0 + 1, x0 + 1) * 0.25f;
            __builtin_nontemporal_store(r, &rois[((size_t)bn * kC + c) * kGP + p]);
        }
    }
}

extern "C" void kernel_launch(void* const* d_in, const int* in_sizes, int n_in,
                              void* d_out, int out_size, void* d_ws, size_t ws_size,
                              hipStream_t stream) {
    const float* images = (const float*)d_in[0];   // [8,32,256,256] f32
    const float* anc    = (const float*)d_in[1];   // [8,600,4] f32
    float* rois    = (float*)d_out;                // [8,600,32,31,31] f32
    float* anc_out = rois + kRoisElems;            // [8,600,2] f32 (tuple output #2)

    const size_t shmem = 2ull * kBufFloats * sizeof(float);  // 160 KB of 320 KB/WGP
    corner_glimpse_kernel<<<dim3(kB * kN), dim3(256), shmem, stream>>>(
        images, anc, rois, anc_out);

    (void)in_sizes; (void)n_in; (void)out_size; (void)d_ws; (void)ws_size;
}